// MLP_2491081032006
// MI455X (gfx1250) — compile-verified
//
#include <hip/hip_runtime.h>
#include <hip/hip_bf16.h>
#include <math.h>

// ---------------------------------------------------------------------------
// Problem constants (from the reference): B=64, I=512, H=1024.
// Dominant cost: plastic_weights is 256MB; read twice + 256MB write => ~33us
// at 23.3 TB/s. GEMMs (~0.7 GFLOP) are done with v_wmma_f32_16x16x4_f32 so we
// keep full f32 precision while using the CDNA5 matrix pipe.
// ---------------------------------------------------------------------------

#define Bdim 64
#define Idim 512
#define Hdim 1024

typedef __attribute__((ext_vector_type(2))) float v2f;
typedef __attribute__((ext_vector_type(4))) float v4f;
typedef __attribute__((ext_vector_type(8))) float v8f;

__device__ __forceinline__ float fast_tanh(float x) {
#if defined(__has_builtin)
#if __has_builtin(__builtin_amdgcn_tanhf)
    return __builtin_amdgcn_tanhf(x);   // gfx1250 V_TANH_F32 (transcendental VALU)
#else
    return tanhf(x);
#endif
#else
    return tanhf(x);
#endif
}

__device__ __forceinline__ float wave_reduce_add(float s) {
    // wave32 butterfly reduction
    #pragma unroll
    for (int o = 16; o > 0; o >>= 1) s += __shfl_xor(s, o, 32);
    return s;
}

// ---------------------------------------------------------------------------
// C(MxN) = act(A(MxK) @ W(NxK)^T + bias), one wave per 16x16 output tile.
// Uses V_WMMA_F32_16X16X4_F32. Fragment layouts (wave32):
//   A 16x4:  v0 = A[m][k0+2*half], v1 = A[m][k0+2*half+1], m = lane%16
//   B 4x16:  v0 = B[k0+2*half][n], v1 = B[k0+2*half+1][n], n = lane%16
//            (B[k][n] = W[n][k] -> contiguous float pair in a W row)
//   C/D 16x16: VGPR r, lanes 0-15 -> M=r, lanes 16-31 -> M=r+8, N = lane%16
// ACT: 0 = none, 1 = tanh
// ---------------------------------------------------------------------------
template <int ACT>
__global__ void gemm_wmma_f32(const float* __restrict__ A,
                              const float* __restrict__ W,
                              const float* __restrict__ bias,
                              float* __restrict__ C,
                              int M, int N, int K) {
    const int lane = threadIdx.x & 31;
    const int half = lane >> 4;       // 0: K pair {0,1}, 1: K pair {2,3}
    const int ml   = lane & 15;       // A row / B col / D col within the tile
    const int n0   = blockIdx.x * 16;
    const int m0   = blockIdx.y * 16;

    const float* __restrict__ arow = A + (size_t)(m0 + ml) * K + 2 * half;
    const float* __restrict__ wrow = W + (size_t)(n0 + ml) * K + 2 * half;

    v8f acc = {};
    for (int k0 = 0; k0 < K; k0 += 4) {
        v2f a, b;
        a.x = arow[k0];
        a.y = arow[k0 + 1];
        b.x = wrow[k0];
        b.y = wrow[k0 + 1];
        acc = __builtin_amdgcn_wmma_f32_16x16x4_f32(
            /*neg_a=*/false, a, /*neg_b=*/false, b,
            /*c_mod=*/(short)0, acc, /*reuse_a=*/false, /*reuse_b=*/false);
    }

    const float bn = bias[n0 + ml];
    #pragma unroll
    for (int r = 0; r < 8; ++r) {
        const int m = m0 + r + 8 * half;
        float v = acc[r] + bn;
        if (ACT == 1) v = fast_tanh(v);
        C[(size_t)m * N + n0 + ml] = v;
    }
}

// ---------------------------------------------------------------------------
// plastic[b][h] = sum_i alpha[h][i] * pw[b][h][i] * x[b][i]
// fused: pre = innate + plastic -> pre_tanh (ws), hidden = tanh(pre) (d_out)
// One wave per (b,h) row; 128-bit vector loads; streams pw once (256MB).
// ---------------------------------------------------------------------------
__global__ void plastic_hidden_kernel(const float* __restrict__ pw,
                                      const float* __restrict__ alpha,
                                      const float* __restrict__ x,
                                      const float* __restrict__ innate,
                                      float* __restrict__ pre_tanh,
                                      float* __restrict__ hidden) {
    const int wid  = blockIdx.x * (blockDim.x >> 5) + (threadIdx.x >> 5);
    const int lane = threadIdx.x & 31;
    if (wid >= Bdim * Hdim) return;
    const int b = wid >> 10;          // wid / H
    const int h = wid & (Hdim - 1);   // wid % H

    const v4f* __restrict__ prow = (const v4f*)(pw + (size_t)wid * Hdim);
    const v4f* __restrict__ arow = (const v4f*)(alpha + (size_t)h * Hdim);
    const v4f* __restrict__ xrow = (const v4f*)(x + (size_t)b * Hdim);

    float s = 0.0f;
    #pragma unroll
    for (int j = lane; j < Hdim / 4; j += 32) {   // 8 iterations
        v4f p = prow[j], al = arow[j], xv = xrow[j];
        s += al.x * p.x * xv.x + al.y * p.y * xv.y +
             al.z * p.z * xv.z + al.w * p.w * xv.w;
    }
    s = wave_reduce_add(s);
    if (lane == 0) {
        float pre = innate[wid] + s;
        pre_tanh[wid] = pre;
        hidden[wid]   = fast_tanh(pre);
    }
}

// hr_in[b][h] = hidden[b][h] + reward[b] * Wr[h] + br[h]
__global__ void hr_in_kernel(const float* __restrict__ hidden,
                             const float* __restrict__ reward,
                             const float* __restrict__ Wr,
                             const float* __restrict__ br,
                             float* __restrict__ hr_in) {
    const int idx = blockIdx.x * blockDim.x + threadIdx.x;
    if (idx >= Bdim * Hdim) return;
    const int b = idx >> 10;
    const int h = idx & (Hdim - 1);
    hr_in[idx] = hidden[idx] + reward[b] * Wr[h] + br[h];
}

// ---------------------------------------------------------------------------
// Per-batch heads: nm = tanh(hrc.Wnm0+b0) - tanh(hrc.Wnm1+b1),
//                  choice = sigmoid(hidden.Wch+bch), value = hidden.Wv+bv
// One wave per batch element; four length-H dots in one lane-strided loop.
// ---------------------------------------------------------------------------
__global__ void heads_kernel(const float* __restrict__ hidden,
                             const float* __restrict__ hrc,
                             const float* __restrict__ Wnm,
                             const float* __restrict__ bnm,
                             const float* __restrict__ Wch,
                             const float* __restrict__ bch,
                             const float* __restrict__ Wv,
                             const float* __restrict__ bv,
                             float* __restrict__ choice_out,
                             float* __restrict__ nm_out,
                             float* __restrict__ value_out,
                             float* __restrict__ nm_ws) {
    const int b    = blockIdx.x;
    const int lane = threadIdx.x & 31;
    const float* __restrict__ hrow = hidden + (size_t)b * Hdim;
    const float* __restrict__ crow = hrc + (size_t)b * Hdim;

    float s0 = 0.f, s1 = 0.f, s2 = 0.f, s3 = 0.f;
    for (int i = lane; i < Hdim; i += 32) {
        const float hv = hrow[i];
        const float cv = crow[i];
        s0 += cv * Wnm[i];
        s1 += cv * Wnm[Hdim + i];
        s2 += hv * Wch[i];
        s3 += hv * Wv[i];
    }
    s0 = wave_reduce_add(s0);
    s1 = wave_reduce_add(s1);
    s2 = wave_reduce_add(s2);
    s3 = wave_reduce_add(s3);
    if (lane == 0) {
        const float nm = fast_tanh(s0 + bnm[0]) - fast_tanh(s1 + bnm[1]);
        nm_out[b] = nm;
        nm_ws[b]  = nm;
        choice_out[b] = 1.0f / (1.0f + __expf(-(s2 + bch[0])));
        value_out[b]  = s3 + bv[0];
    }
}

// ---------------------------------------------------------------------------
// new_pw[b][h][i] = clip(pw + nm[b] * tanh(pre[b][h] * x[b][i]), +-50)
// One wave per (b,h) row. pw load is regular-temporal (hits L2 lines left by
// plastic_hidden_kernel); new_pw store is NON-temporal (write-once stream).
// ---------------------------------------------------------------------------
__global__ void update_pw_kernel(const float* __restrict__ pw,
                                 const float* __restrict__ pre_tanh,
                                 const float* __restrict__ x,
                                 const float* __restrict__ nm_ws,
                                 float* __restrict__ new_pw) {
    const int wid  = blockIdx.x * (blockDim.x >> 5) + (threadIdx.x >> 5);
    const int lane = threadIdx.x & 31;
    if (wid >= Bdim * Hdim) return;
    const int b = wid >> 10;

    const float pre = pre_tanh[wid];
    const float nm  = nm_ws[b];

    const v4f* __restrict__ prow = (const v4f*)(pw + (size_t)wid * Hdim);
    const v4f* __restrict__ xrow = (const v4f*)(x + (size_t)b * Hdim);
    v4f* __restrict__ orow       = (v4f*)(new_pw + (size_t)wid * Hdim);

    #pragma unroll
    for (int j = lane; j < Hdim / 4; j += 32) {   // 8 iterations
        v4f p  = prow[j];
        v4f xv = xrow[j];
        v4f o;
        o.x = fminf(fmaxf(p.x + nm * fast_tanh(pre * xv.x), -50.0f), 50.0f);
        o.y = fminf(fmaxf(p.y + nm * fast_tanh(pre * xv.y), -50.0f), 50.0f);
        o.z = fminf(fmaxf(p.z + nm * fast_tanh(pre * xv.z), -50.0f), 50.0f);
        o.w = fminf(fmaxf(p.w + nm * fast_tanh(pre * xv.w), -50.0f), 50.0f);
        __builtin_nontemporal_store(o, &orow[j]);
    }
}

// ---------------------------------------------------------------------------
// Host-side launcher
// ---------------------------------------------------------------------------
extern "C" void kernel_launch(void* const* d_in, const int* in_sizes, int n_in,
                              void* d_out, int out_size, void* d_ws, size_t ws_size,
                              hipStream_t stream) {
    // inputs (setup_inputs dict order)
    const float* items  = (const float*)d_in[0];   // (B,I)
    const float* pw     = (const float*)d_in[1];   // (B,H,H)
    const float* reward = (const float*)d_in[2];   // (B,1)
    const float* We     = (const float*)d_in[3];   // (H,I)
    const float* be     = (const float*)d_in[4];
    const float* W1     = (const float*)d_in[5];
    const float* b1     = (const float*)d_in[6];
    const float* W2     = (const float*)d_in[7];
    const float* b2     = (const float*)d_in[8];
    const float* Wfc2   = (const float*)d_in[9];
    const float* bfc2   = (const float*)d_in[10];
    const float* Whr    = (const float*)d_in[11];
    const float* bhr    = (const float*)d_in[12];
    const float* Wch    = (const float*)d_in[13];  // (1,H)
    const float* bch    = (const float*)d_in[14];  // (1,)
    const float* Wr     = (const float*)d_in[15];  // (H,1)
    const float* br     = (const float*)d_in[16];  // (H,)
    const float* Wnm    = (const float*)d_in[17];  // (2,H)
    const float* bnm    = (const float*)d_in[18];  // (2,)
    const float* alpha  = (const float*)d_in[19];  // (H,H)
    const float* Wv     = (const float*)d_in[20];  // (1,H)
    const float* bv     = (const float*)d_in[21];  // (1,)

    // outputs, concatenated flat: choice(64), nm(64), value(64), new_pw(64M), hidden(64K)
    float* out        = (float*)d_out;
    float* choice_out = out;
    float* nm_out     = out + 64;
    float* value_out  = out + 128;
    float* new_pw     = out + 192;
    float* hidden     = out + 192 + (size_t)Bdim * Hdim * Hdim;

    // workspace layout (floats)
    float* ws       = (float*)d_ws;
    float* a0       = ws;                        // (B,H) after embedding+tanh
    float* a1       = a0 + Bdim * Hdim;          // after layer1
    float* a2       = a1 + Bdim * Hdim;          // after layer2 ("x")
    float* innate   = a2 + Bdim * Hdim;
    float* pre_tanh = innate + Bdim * Hdim;
    float* hr_in    = pre_tanh + Bdim * Hdim;
    float* hrc      = hr_in + Bdim * Hdim;
    float* nm_ws    = hrc + Bdim * Hdim;         // (B,)

    const dim3 gtile(Hdim / 16, Bdim / 16);      // (64, 4) tiles, 1 wave each
    const dim3 wblk(32);

    // stacked Linear + tanh
    gemm_wmma_f32<1><<<gtile, wblk, 0, stream>>>(items, We, be, a0, Bdim, Hdim, Idim);
    gemm_wmma_f32<1><<<gtile, wblk, 0, stream>>>(a0, W1, b1, a1, Bdim, Hdim, Hdim);
    gemm_wmma_f32<1><<<gtile, wblk, 0, stream>>>(a1, W2, b2, a2, Bdim, Hdim, Hdim);
    // innate (no activation)
    gemm_wmma_f32<0><<<gtile, wblk, 0, stream>>>(a2, Wfc2, bfc2, innate, Bdim, Hdim, Hdim);

    // plastic reduction + hidden   (one wave per (b,h) row; 8 waves/block)
    plastic_hidden_kernel<<<(Bdim * Hdim) / 8, 256, 0, stream>>>(
        pw, alpha, a2, innate, pre_tanh, hidden);

    // hidden + reward embedding
    hr_in_kernel<<<(Bdim * Hdim) / 256, 256, 0, stream>>>(hidden, reward, Wr, br, hr_in);

    // hrc = tanh(hr_in @ Whr^T + bhr)
    gemm_wmma_f32<1><<<gtile, wblk, 0, stream>>>(hr_in, Whr, bhr, hrc, Bdim, Hdim, Hdim);

    // per-batch heads: nm / choice / value
    heads_kernel<<<Bdim, wblk, 0, stream>>>(hidden, hrc, Wnm, bnm, Wch, bch, Wv, bv,
                                            choice_out, nm_out, value_out, nm_ws);

    // streaming plastic-weight update (dominant 512MB of traffic)
    update_pw_kernel<<<(Bdim * Hdim) / 8, 256, 0, stream>>>(
        pw, pre_tanh, a2, nm_ws, new_pw);
}